// Group_19361712570464
// MI455X (gfx1250) — compile-verified
//
#include <hip/hip_runtime.h>

#define BATCH 16
#define NPTS  16384
#define NGRP  512
#define GSZ   32

typedef float v2f __attribute__((ext_vector_type(2)));
typedef float v8f __attribute__((ext_vector_type(8)));

// ---------------------------------------------------------------------------
// Kernel 1: Farthest Point Sampling. One block per batch, 1024 threads
// (32 wave32 waves). Each thread owns 16 points in registers (coords + running
// min-dist), per-step argmax via wave shfl_xor reduction then a 32-entry LDS
// reduction by wave 0. Matches jnp.argmax first-occurrence (smallest index on
// ties). Writes centers (B,G,3) straight into the tail of d_out.
// ---------------------------------------------------------------------------
#define FPS_T 1024
#define PPT   (NPTS / FPS_T)   // 16 points per thread

__global__ __launch_bounds__(FPS_T)
void fps_kernel(const float* __restrict__ xyz, float* __restrict__ centers) {
  const int b = blockIdx.x;
  const int t = threadIdx.x;
  const float* __restrict__ xb = xyz + (size_t)b * NPTS * 3;

  float cx[PPT], cy[PPT], cz[PPT], md[PPT];
#pragma unroll
  for (int j = 0; j < PPT; ++j) {
    int n = t + j * FPS_T;                 // strided ownership, coalesced loads
    cx[j] = xb[n * 3 + 0];
    cy[j] = xb[n * 3 + 1];
    cz[j] = xb[n * 3 + 2];
    md[j] = __builtin_inff();
  }

  __shared__ float sP[3];
  __shared__ float wVal[32];
  __shared__ int   wIdx[32];

  if (t == 0) {
    float x0 = xb[0], y0 = xb[1], z0 = xb[2];   // scan emits index 0 first
    sP[0] = x0; sP[1] = y0; sP[2] = z0;
    float* c0 = centers + (size_t)b * NGRP * 3;
    c0[0] = x0; c0[1] = y0; c0[2] = z0;
  }

  for (int s = 1; s < NGRP; ++s) {
    __syncthreads();
    const float qx = sP[0], qy = sP[1], qz = sP[2];
    float bv = -1.0f;
    int   bi = 0;
#pragma unroll
    for (int j = 0; j < PPT; ++j) {
      float dx = cx[j] - qx, dy = cy[j] - qy, dz = cz[j] - qz;
      float d  = __builtin_fmaf(dz, dz, __builtin_fmaf(dy, dy, dx * dx));
      float m  = __builtin_fminf(md[j], d);
      md[j] = m;
      if (m > bv) { bv = m; bi = t + j * FPS_T; }  // strict > keeps lowest index
    }
#pragma unroll
    for (int off = 16; off > 0; off >>= 1) {
      float ov = __shfl_xor(bv, off, 32);
      int   oi = __shfl_xor(bi, off, 32);
      if (ov > bv || (ov == bv && oi < bi)) { bv = ov; bi = oi; }
    }
    if ((t & 31) == 0) { wVal[t >> 5] = bv; wIdx[t >> 5] = bi; }
    __syncthreads();
    if (t < 32) {
      bv = wVal[t];
      bi = wIdx[t];
#pragma unroll
      for (int off = 16; off > 0; off >>= 1) {
        float ov = __shfl_xor(bv, off, 32);
        int   oi = __shfl_xor(bi, off, 32);
        if (ov > bv || (ov == bv && oi < bi)) { bv = ov; bi = oi; }
      }
      if (t == 0) {
        float x0 = xb[bi * 3 + 0], y0 = xb[bi * 3 + 1], z0 = xb[bi * 3 + 2];
        sP[0] = x0; sP[1] = y0; sP[2] = z0;
        float* cs = centers + ((size_t)b * NGRP + s) * 3;
        cs[0] = x0; cs[1] = y0; cs[2] = z0;
      }
    }
  }
}

// ---------------------------------------------------------------------------
// Kernel 2: KNN (top-32 of squared distance) + gather + recenter.
// One wave per 16 centers. d2 = |c|^2 - 2 c.x + |x|^2; the -2 c.x Gram tiles
// come from two V_WMMA_F32_16X16X4_F32 per 32-point chunk (-2*c folded into A,
// K padded to 4). Full d2 tiles land in a padded LDS tile (row stride 36
// floats: 16B-aligned rows, conflict-free b128 row reads). Lanes 0-15 then
// pull their center's 32 values with 8 independent b128 loads, build a
// candidate bitmask vs the running 32nd-smallest, and only run the O(32)
// insert body for actual candidates (ctz-driven sparse loop). The top-32 list
// uses value-match replacement + v_max rescan, seeded with distinct huge
// sentinels so the fill phase needs no special casing.
// ---------------------------------------------------------------------------
#define KNN_WPB 8
#define CH 32                       // points per chunk (2 WMMA subtiles)
#define TPAD 36                     // padded row length (floats)

__global__ __launch_bounds__(KNN_WPB * 32)
void knn_kernel(const float* __restrict__ xyz, const float* __restrict__ centers,
                float* __restrict__ nei) {
  __shared__ __align__(16) float tile[KNN_WPB][16][TPAD];
  const int lane = threadIdx.x & 31;
  const int wv   = threadIdx.x >> 5;
  const int gw   = blockIdx.x * KNN_WPB + wv;
  const int b    = gw >> 5;                 // 32 waves per batch
  const int m0   = (gw & 31) << 4;          // 16 centers per wave
  const float* __restrict__ xb = xyz + (size_t)b * NPTS * 3;
  const float* __restrict__ cb = centers + ((size_t)b * NGRP + m0) * 3;

  const int hh = lane >> 4;                 // lane half: 0 -> K0,K1 ; 1 -> K2,K3
  const int lm = lane & 15;

  const float ccx = cb[lm * 3 + 0];
  const float ccy = cb[lm * 3 + 1];
  const float ccz = cb[lm * 3 + 2];

  v2f A;
  A.x = -2.0f * (hh ? ccz : ccx);
  A.y = hh ? 0.0f : (-2.0f * ccy);

  float cn2[16];                            // |c|^2, replicated across lanes
#pragma unroll
  for (int j = 0; j < 16; ++j) {
    float x0 = cb[j * 3 + 0], y0 = cb[j * 3 + 1], z0 = cb[j * 3 + 2];
    cn2[j] = __builtin_fmaf(z0, z0, __builtin_fmaf(y0, y0, x0 * x0));
  }

  // top-32 list: distinct huge sentinels (all >> any real d^2, all distinct in
  // f32 since 1e24 > ulp(1e30)) make value-match replacement safe during fill.
  float ld[GSZ];
  int   li[GSZ];
#pragma unroll
  for (int k = 0; k < GSZ; ++k) { ld[k] = 1.0e30f + (float)k * 1.0e24f; li[k] = 0; }
  float curMax = 1.0e30f + 31.0f * 1.0e24f;

  for (int n0 = 0; n0 < NPTS; n0 += CH) {
    const int na = n0 + lm;                 // cols 0-15
    const int nb = n0 + 16 + lm;            // cols 16-31
    float ax = xb[na * 3 + 0], ay = xb[na * 3 + 1], az = xb[na * 3 + 2];
    float bx = xb[nb * 3 + 0], by = xb[nb * 3 + 1], bz = xb[nb * 3 + 2];
    float xa2 = __builtin_fmaf(az, az, __builtin_fmaf(ay, ay, ax * ax));
    float xb2 = __builtin_fmaf(bz, bz, __builtin_fmaf(by, by, bx * bx));

    v2f B0, B1;
    B0.x = hh ? az : ax;  B0.y = hh ? 0.0f : ay;
    B1.x = hh ? bz : bx;  B1.y = hh ? 0.0f : by;

    v8f acc0 = {}, acc1 = {};
    acc0 = __builtin_amdgcn_wmma_f32_16x16x4_f32(false, A, false, B0,
                                                 (short)0, acc0, false, false);
    acc1 = __builtin_amdgcn_wmma_f32_16x16x4_f32(false, A, false, B1,
                                                 (short)0, acc1, false, false);

#pragma unroll
    for (int j = 0; j < 8; ++j) {
      float c2 = hh ? cn2[j + 8] : cn2[j];
      const int row = j + (hh << 3);
      tile[wv][row][lm]      = acc0[j] + xa2 + c2;   // full d^2, cols 0-15
      tile[wv][row][lm + 16] = acc1[j] + xb2 + c2;   // cols 16-31
    }

    __builtin_amdgcn_wave_barrier();   // per-wave DS ops are in-order; just
    asm volatile("" ::: "memory");     // stop the compiler reordering them

    if (lane < 16) {                   // lane == center row
      const float4* __restrict__ rp = (const float4*)&tile[wv][lane][0];
      unsigned mask = 0u;
#pragma unroll
      for (int j = 0; j < 8; ++j) {    // 8 independent b128 row loads
        float4 r = rp[j];
        mask |= (r.x < curMax) ? (1u << (4 * j + 0)) : 0u;
        mask |= (r.y < curMax) ? (1u << (4 * j + 1)) : 0u;
        mask |= (r.z < curMax) ? (1u << (4 * j + 2)) : 0u;
        mask |= (r.w < curMax) ? (1u << (4 * j + 3)) : 0u;
      }
      while (mask) {                   // sparse: usually 0 iterations
        const int q = __builtin_ctz(mask);
        mask &= mask - 1u;
        float d = tile[wv][lane][q];
        if (d < curMax) {              // recheck: curMax shrinks within chunk
          const int pi = n0 + q;
#pragma unroll
          for (int k = 0; k < GSZ; ++k) {   // value-match replace of the max
            bool eq = (ld[k] == curMax);
            ld[k] = eq ? d : ld[k];
            li[k] = eq ? pi : li[k];
          }
          float nm = -__builtin_inff();
#pragma unroll
          for (int k = 0; k < GSZ; ++k) nm = __builtin_fmaxf(nm, ld[k]);
          curMax = nm;
        }
      }
    }

    __builtin_amdgcn_wave_barrier();
    asm volatile("" ::: "memory");
  }

  // Emit in ascending-distance order (ties -> lower index), matching top_k.
  if (lane < 16) {
    const int m = m0 + lm;
    float* __restrict__ out = nei + (((size_t)b * NGRP + m) * GSZ) * 3;
    unsigned used = 0u;
#pragma unroll 1
    for (int o = 0; o < GSZ; ++o) {
      float bd = __builtin_inff();
      int   bi2 = 0x7fffffff;
      int   bp = 0;
#pragma unroll
      for (int k = 0; k < GSZ; ++k) {
        bool  u = (used >> k) & 1u;
        float d = u ? __builtin_inff() : ld[k];
        int   i2 = li[k];
        if (d < bd || (d == bd && i2 < bi2)) { bd = d; bi2 = i2; bp = k; }
      }
      used |= (1u << bp);
      out[o * 3 + 0] = xb[bi2 * 3 + 0] - ccx;
      out[o * 3 + 1] = xb[bi2 * 3 + 1] - ccy;
      out[o * 3 + 2] = xb[bi2 * 3 + 2] - ccz;
    }
  }
}

// ---------------------------------------------------------------------------
extern "C" void kernel_launch(void* const* d_in, const int* in_sizes, int n_in,
                              void* d_out, int out_size, void* d_ws, size_t ws_size,
                              hipStream_t stream) {
  (void)in_sizes; (void)n_in; (void)out_size; (void)d_ws; (void)ws_size;
  const float* xyz = (const float*)d_in[0];
  float* out = (float*)d_out;
  float* nei     = out;                                      // (B,G,32,3)
  float* centers = out + (size_t)BATCH * NGRP * GSZ * 3;     // (B,G,3)

  fps_kernel<<<BATCH, FPS_T, 0, stream>>>(xyz, centers);
  knn_kernel<<<(BATCH * (NGRP / 16)) / KNN_WPB, KNN_WPB * 32, 0, stream>>>(
      xyz, centers, nei);
}